// PointNetSetAbstraction_16758962389302
// MI455X (gfx1250) — compile-verified
//
#include <hip/hip_runtime.h>
#include <math.h>

// ---------------- problem constants ----------------
constexpr int BB  = 8;
constexpr int NN  = 4096;
constexpr int NP  = 1024;
constexpr int NSAMP = 32;
constexpr int CF  = 64;           // input feature channels
constexpr int C0P = 68;           // 3+CF=67 padded to multiple of 4
constexpr int C0R = 67;
constexpr int O0  = 64, O1 = 128, O2 = 256;
constexpr int M   = BB * NP * NSAMP;   // 262144 GEMM rows
constexpr int MT  = M / 16;            // 16384 row-tiles
constexpr float EPS = 1e-5f;

typedef float v2f __attribute__((ext_vector_type(2)));
typedef float v8f __attribute__((ext_vector_type(8)));

// ---------------------------------------------------------------------------
// Activation layout everywhere: X[C/4][M][4]  (channel quads innermost).
// A-fragment for WMMA f32 16x16x4 is then a contiguous float2 per lane
// (one global_load_b64, coalesced), instead of two stride-M b32 loads.
// ---------------------------------------------------------------------------

// Kernel 1: gather + concat into quad-interleaved X0; also write new_xyz.
__global__ __launch_bounds__(256)
void build_x0(const float* __restrict__ xyz, const float* __restrict__ pts,
              const int* __restrict__ sidx, const int* __restrict__ nidx,
              float* __restrict__ X0, float* __restrict__ new_xyz)
{
    int m = blockIdx.x * 256 + threadIdx.x;
    int s = m & (NSAMP - 1);
    int p = (m / NSAMP) & (NP - 1);
    int b = m / (NSAMP * NP);

    int ni = nidx[m];
    int si = sidx[b * NP + p];
    const float* xb = xyz + (size_t)b * NN * 3;
    float sx = xb[si * 3 + 0], sy = xb[si * 3 + 1], sz = xb[si * 3 + 2];
    float rel[3];
    rel[0] = xb[ni * 3 + 0] - sx;
    rel[1] = xb[ni * 3 + 1] - sy;
    rel[2] = xb[ni * 3 + 2] - sz;

    const float* pb = pts + (size_t)b * CF * NN;
#pragma unroll
    for (int q = 0; q < C0P / 4; ++q) {      // 17 channel quads
        float4 t;
        float v[4];
#pragma unroll
        for (int j = 0; j < 4; ++j) {
            int c = 4 * q + j;
            v[j] = (c < 3) ? rel[c]
                 : (c < C0R) ? pb[(size_t)(c - 3) * NN + ni]
                 : 0.0f;                      // zero pad channel 67
        }
        t.x = v[0]; t.y = v[1]; t.z = v[2]; t.w = v[3];
        ((float4*)X0)[(size_t)q * M + m] = t; // coalesced 16B per lane
    }

    if (s == 0) {
        float* o = new_xyz + (size_t)(b * NP + p) * 3;
        o[0] = sx; o[1] = sy; o[2] = sz;
    }
}

// ---------------------------------------------------------------------------
// GEMM: Y = W * X on V_WMMA_F32_16X16X4_F32, quad-interleaved activations.
// Block = 256 threads = 8 waves; each wave owns NT=4 16x16 output tiles that
// share one A fragment per K-step (1 b64 A-load feeds 4 wmma).
//
// Fragment layouts per ISA 7.12.2 (f32):
//   A 16x4 : lanes 0-15 -> row M=lane, vgpr0=K0 vgpr1=K1; lanes 16-31 -> K2,K3
//   B 4x16 : mirrored per column N (lane = N, halves give K pairs)
//   C/D    : vgpr v -> row v (lanes 0-15) / row v+8 (lanes 16-31), N = lane&15
//
// MODE 0: write Y (quad-interleaved, coalesced) + per-tile sum/sumsq partials.
// MODE 1: fused final layer — sum/sumsq + max/min partials, no Y store.
// Bias omitted: it cancels exactly in training-mode BatchNorm.
// ---------------------------------------------------------------------------
template <int CIN, int CINR, int COUT, int MODE, int NT>
__global__ __launch_bounds__(256)
void gemm_wmma(const float* __restrict__ X, const float* __restrict__ W,
               float* __restrict__ Y,
               float* __restrict__ pS, float* __restrict__ pQ,
               float* __restrict__ pMax, float* __restrict__ pMin)
{
    constexpr int WS = CIN + 1;                 // padded LDS stride
    __shared__ float Wlds[NT * 16 * WS];
    __shared__ float tile[8][16][17];

    const int tid    = threadIdx.x;
    const int nBase0 = blockIdx.y * (16 * NT);

    // Stage NT*16 x CIN weight tile (zero-fill padded K columns)
    for (int i = tid; i < NT * 16 * CIN; i += 256) {
        int n = i / CIN, k = i - n * CIN;
        Wlds[n * WS + k] = (k < CINR) ? W[(size_t)(nBase0 + n) * CINR + k] : 0.0f;
    }
    __syncthreads();

    const int wave  = tid >> 5;
    const int lane  = tid & 31;
    const int lh    = lane & 15;
    const bool hi   = lane >= 16;
    const int mTile = blockIdx.x * 8 + wave;
    const int mBase = mTile * 16;

    v8f acc[NT];
#pragma unroll
    for (int nt = 0; nt < NT; ++nt) acc[nt] = (v8f){};

    // A pointer: quad-interleaved -> lane's K-pair is contiguous float2
    const float* abase = X + ((size_t)(mBase + lh)) * 4 + (hi ? 2 : 0);

#pragma unroll
    for (int k4 = 0; k4 < CIN; k4 += 4) {
        const int kA = k4 + (hi ? 2 : 0);
        v2f a = *(const v2f*)(abase + (size_t)(k4 >> 2) * M * 4);
#pragma unroll
        for (int nt = 0; nt < NT; ++nt) {
            const float* wrow = Wlds + (nt * 16 + lh) * WS;
            v2f bf; bf.x = wrow[kA]; bf.y = wrow[kA + 1];
            acc[nt] = __builtin_amdgcn_wmma_f32_16x16x4_f32(
                          false, a, false, bf, (short)0, acc[nt], false, false);
        }
    }

    float (*tp)[17] = tile[wave];
#pragma unroll
    for (int nt = 0; nt < NT; ++nt) {
        const int nBase = nBase0 + nt * 16;

        // Scatter accumulator into LDS tile[m][n]
#pragma unroll
        for (int v = 0; v < 8; ++v)
            tp[v + (hi ? 8 : 0)][lh] = acc[nt][v];
        __syncthreads();

        // Per-tile column (channel) reductions over the 16 rows.
        float s = 0.f, q = 0.f, mx = -3.402823466e38f, mn = 3.402823466e38f;
#pragma unroll
        for (int r = 0; r < 16; ++r) {
            float x = tp[r][lh];
            s += x; q += x * x;
            mx = fmaxf(mx, x); mn = fminf(mn, x);
        }
        const size_t ch = (size_t)(nBase + lh);
        if (!hi) {
            pS[ch * MT + mTile] = s;
            if (MODE == 1) pMax[ch * MT + mTile] = mx;
        } else {
            pQ[ch * MT + mTile] = q;
            if (MODE == 1) pMin[ch * MT + mTile] = mn;
        }

        if (MODE == 0) {
            // Quad-interleaved Y store: lane -> row m=lane&15, quad = lane>>4
            const int mrow = lane & 15;
            const int qsel = lane >> 4;
#pragma unroll
            for (int h = 0; h < 2; ++h) {
                const int nb4 = qsel + 2 * h;   // channel quad within tile
                float4 t;
                t.x = tp[mrow][nb4 * 4 + 0];
                t.y = tp[mrow][nb4 * 4 + 1];
                t.z = tp[mrow][nb4 * 4 + 2];
                t.w = tp[mrow][nb4 * 4 + 3];
                ((float4*)Y)[(size_t)((nBase >> 2) + nb4) * M + mBase + mrow] = t;
            }
        }
        __syncthreads();                        // tile reused by next nt
    }
}

// ---------------------------------------------------------------------------
// Deterministic per-channel reduction of the MT tile partials -> BN affine.
// ---------------------------------------------------------------------------
__global__ __launch_bounds__(256)
void stats_reduce(const float* __restrict__ pS, const float* __restrict__ pQ,
                  const float* __restrict__ gamma, const float* __restrict__ beta,
                  float* __restrict__ sc, float* __restrict__ sh)
{
    __shared__ float aS[256], aQ[256];
    const int c = blockIdx.x, t = threadIdx.x;
    float s = 0.f, q = 0.f;
    for (int i = t; i < MT; i += 256) {
        s += pS[(size_t)c * MT + i];
        q += pQ[(size_t)c * MT + i];
    }
    aS[t] = s; aQ[t] = q;
    __syncthreads();
    for (int st = 128; st > 0; st >>= 1) {
        if (t < st) { aS[t] += aS[t + st]; aQ[t] += aQ[t + st]; }
        __syncthreads();
    }
    if (t == 0) {
        float mean = aS[0] / (float)M;
        float var  = aQ[0] / (float)M - mean * mean;
        float inv  = rsqrtf(var + EPS);
        float a    = gamma[c] * inv;
        sc[c] = a;
        sh[c] = beta[c] - mean * a;
    }
}

// In-place BN + ReLU over quad-interleaved Y[COUT/4][M][4], float4 vectorized.
__global__ __launch_bounds__(256)
void bn_relu(float* __restrict__ Y, const float* __restrict__ sc,
             const float* __restrict__ sh)
{
    int f = blockIdx.x * 256 + threadIdx.x;        // float4 index = cblk*M + m
    int cb = (f / M) * 4;                          // base channel of this quad
    float4 v = ((float4*)Y)[f];
    v.x = fmaxf(fmaf(sc[cb + 0], v.x, sh[cb + 0]), 0.f);
    v.y = fmaxf(fmaf(sc[cb + 1], v.y, sh[cb + 1]), 0.f);
    v.z = fmaxf(fmaf(sc[cb + 2], v.z, sh[cb + 2]), 0.f);
    v.w = fmaxf(fmaf(sc[cb + 3], v.w, sh[cb + 3]), 0.f);
    ((float4*)Y)[f] = v;
}

// Final output: combine the two s-half tiles' max/min, apply BN affine with
// the sign rule (affine monotone: scale>=0 -> max, else min), then ReLU.
// Output layout: new_points (B, O2, NP) row-major.
__global__ __launch_bounds__(256)
void finalize(const float* __restrict__ pMax, const float* __restrict__ pMin,
              const float* __restrict__ sc, const float* __restrict__ sh,
              float* __restrict__ out)
{
    int idx = blockIdx.x * 256 + threadIdx.x;      // b*O2*NP total
    int p = idx & (NP - 1);
    int c = (idx / NP) & (O2 - 1);
    int b = idx / (NP * O2);
    int t0 = (b * NP + p) * 2;                     // mTile = (b*NP+p)*2 + s/16
    float mx = fmaxf(pMax[(size_t)c * MT + t0], pMax[(size_t)c * MT + t0 + 1]);
    float mn = fminf(pMin[(size_t)c * MT + t0], pMin[(size_t)c * MT + t0 + 1]);
    float a = sc[c];
    float y = (a >= 0.f ? a * mx : a * mn) + sh[c];
    out[idx] = fmaxf(y, 0.f);
}

// ---------------------------------------------------------------------------
extern "C" void kernel_launch(void* const* d_in, const int* in_sizes, int n_in,
                              void* d_out, int out_size, void* d_ws, size_t ws_size,
                              hipStream_t stream)
{
    const float* xyz  = (const float*)d_in[0];
    const float* pts  = (const float*)d_in[1];
    const int*   sidx = (const int*)d_in[2];
    const int*   nidx = (const int*)d_in[3];
    const float* W0 = (const float*)d_in[4];
    const float* g0 = (const float*)d_in[6];
    const float* be0 = (const float*)d_in[7];
    const float* W1 = (const float*)d_in[8];
    const float* g1 = (const float*)d_in[10];
    const float* be1 = (const float*)d_in[11];
    const float* W2 = (const float*)d_in[12];
    const float* g2 = (const float*)d_in[14];
    const float* be2 = (const float*)d_in[15];

    float* out = (float*)d_out;                    // [0,24576): new_xyz; rest: new_points

    // Workspace layout (floats); peak ~268 MB, R1 reused X0 -> Y1.
    float* R1   = (float*)d_ws;                    // 128*M  (X0 uses first 68*M)
    float* R2   = R1 + (size_t)O1 * M;             // 64*M   (Y0)
    float* pS   = R2 + (size_t)O0 * M;             // 256*MT
    float* pQ   = pS + (size_t)O2 * MT;
    float* pMax = pQ + (size_t)O2 * MT;
    float* pMin = pMax + (size_t)O2 * MT;
    float* sc   = pMin + (size_t)O2 * MT;          // 256
    float* sh   = sc + O2;                         // 256

    // 1) gather/concat + new_xyz
    build_x0<<<M / 256, 256, 0, stream>>>(xyz, pts, sidx, nidx, R1, out);

    // 2) layer 0: X0(68ch) -> Y0(64ch); 4 N-tiles/wave -> grid.y = 1
    gemm_wmma<C0P, C0R, O0, 0, 4><<<dim3(M / 128, O0 / 64), 256, 0, stream>>>(
        R1, W0, R2, pS, pQ, nullptr, nullptr);
    stats_reduce<<<O0, 256, 0, stream>>>(pS, pQ, g0, be0, sc, sh);
    bn_relu<<<(size_t)O0 * M / 1024, 256, 0, stream>>>(R2, sc, sh);

    // 3) layer 1: Y0 -> Y1 (reuses R1)
    gemm_wmma<O0, O0, O1, 0, 4><<<dim3(M / 128, O1 / 64), 256, 0, stream>>>(
        R2, W1, R1, pS, pQ, nullptr, nullptr);
    stats_reduce<<<O1, 256, 0, stream>>>(pS, pQ, g1, be1, sc, sh);
    bn_relu<<<(size_t)O1 * M / 1024, 256, 0, stream>>>(R1, sc, sh);

    // 4) layer 2 fused: Y1 -> stats + per-tile max/min (no 268 MB Y2)
    gemm_wmma<O1, O1, O2, 1, 4><<<dim3(M / 128, O2 / 64), 256, 0, stream>>>(
        R1, W2, nullptr, pS, pQ, pMax, pMin);
    stats_reduce<<<O2, 256, 0, stream>>>(pS, pQ, g2, be2, sc, sh);

    // 5) BN affine + NS-max + ReLU -> new_points
    finalize<<<(BB * O2 * NP) / 256, 256, 0, stream>>>(
        pMax, pMin, sc, sh, out + (size_t)BB * NP * 3);
}